// CPMLoss_12326556139945
// MI455X (gfx1250) — compile-verified
//
#include <hip/hip_runtime.h>
#include <math.h>

// Shapes from the reference
#define B_    512
#define K_    8
#define P_    6
#define D_    1024
#define NIDS  64
#define MARGIN_ 0.2f

typedef float v2f __attribute__((ext_vector_type(2)));
typedef float v8f __attribute__((ext_vector_type(8)));

// ---------------------------------------------------------------------------
// Kernel 1: per-id modality counts + number of valid ids
// ---------------------------------------------------------------------------
__global__ void cpm_counts(const int* __restrict__ pids,
                           const int* __restrict__ camids,
                           float* __restrict__ cnt_rgb,
                           float* __restrict__ cnt_sar,
                           float* __restrict__ id_count) {
    __shared__ float validf[NIDS];
    const int id = threadIdx.x;                 // blockDim.x == 64
    int cr = 0, cs = 0;
    for (int b = 0; b < B_; ++b) {
        if (pids[b] == id) {
            if (camids[b] == 0) cr++; else cs++;
        }
    }
    cnt_rgb[id] = (float)cr;
    cnt_sar[id] = (float)cs;
    validf[id]  = (cr > 0 && cs > 0) ? 1.0f : 0.0f;
    __syncthreads();
    if (id == 0) {
        float s = 0.0f;
        for (int i = 0; i < NIDS; ++i) s += validf[i];
        *id_count = s;
    }
}

// ---------------------------------------------------------------------------
// Kernel 2: modality centers, one block per (id, p). Scalar pid pre-check so
// each block only streams rows of its own id (total HBM read = |f_original|).
// ---------------------------------------------------------------------------
__global__ void cpm_centers(const float* __restrict__ f_orig,
                            const int*   __restrict__ pids,
                            const int*   __restrict__ camids,
                            const float* __restrict__ cnt_rgb,
                            const float* __restrict__ cnt_sar,
                            float* __restrict__ center_rgb,
                            float* __restrict__ center_sar) {
    __shared__ int sp[B_];
    __shared__ int sc[B_];
    const int id = blockIdx.x / P_;
    const int p  = blockIdx.x % P_;
    for (int i = threadIdx.x; i < B_; i += blockDim.x) {
        sp[i] = pids[i];
        sc[i] = camids[i];
    }
    __syncthreads();

    const int d = threadIdx.x * 4;              // blockDim.x == 256 -> covers D
    float arx = 0.f, ary = 0.f, arz = 0.f, arw = 0.f;
    float asx = 0.f, asy = 0.f, asz = 0.f, asw = 0.f;
    for (int b = 0; b < B_; ++b) {
        if (sp[b] != id) continue;              // uniform (scalar) skip
        const float4 v = *(const float4*)(f_orig + ((size_t)b * P_ + p) * D_ + d);
        if (sc[b] == 0) { arx += v.x; ary += v.y; arz += v.z; arw += v.w; }
        else            { asx += v.x; asy += v.y; asz += v.z; asw += v.w; }
    }
    float cr = cnt_rgb[id]; cr = cr > 1.f ? cr : 1.f;
    float cs = cnt_sar[id]; cs = cs > 1.f ? cs : 1.f;
    const size_t base = ((size_t)id * P_ + p) * D_ + d;
    float4 outr; outr.x = arx / cr; outr.y = ary / cr; outr.z = arz / cr; outr.w = arw / cr;
    float4 outs; outs.x = asx / cs; outs.y = asy / cs; outs.z = asz / cs; outs.w = asw / cs;
    *(float4*)(center_rgb + base) = outr;
    *(float4*)(center_sar + base) = outs;
}

// ---------------------------------------------------------------------------
// Kernel 3: per-sample weight w[b] = valid ? 1/(max(grp,1)*K) : 0
// ---------------------------------------------------------------------------
__global__ void cpm_weights(const int* __restrict__ pids,
                            const int* __restrict__ camids,
                            const float* __restrict__ cnt_rgb,
                            const float* __restrict__ cnt_sar,
                            float* __restrict__ wsamp) {
    const int b = blockIdx.x * blockDim.x + threadIdx.x;
    if (b >= B_) return;
    const int  pid = pids[b];
    const bool rgb = (camids[b] == 0);
    const float grp = rgb ? cnt_rgb[pid] : cnt_sar[pid];
    const bool valid = (cnt_rgb[pid] > 0.f) && (cnt_sar[pid] > 0.f);
    const float g = grp > 1.f ? grp : 1.f;
    wsamp[b] = valid ? 1.0f / (g * (float)K_) : 0.0f;
}

// ---------------------------------------------------------------------------
// Kernel 4 (main, WMMA): one wave32 per (b,p).
//   A rows (M): 0..7 = g_k, 8 = cross-center c, 9 = f_original o,
//               10..15 = DON'T-CARE (loaded from a valid dummy address)
//   B cols (N): 0 = c, 1 = o, 2..9 = g_k, 10..15 = DON'T-CARE
//   Accumulate over D_=1024 in K=4 chunks via V_WMMA_F32_16X16X4_F32:
//     D[k][0]=g_k.c  D[k][1]=g_k.o  D[k][2+k]=|g_k|^2  D[8][0]=|c|^2  D[9][1]=|o|^2
//   Don't-care rows/cols only touch D entries we never read, so the inner
//   loop is branch-free: 4 global_load_b64 + 2 WMMAs (two accumulators to
//   break the D->C dependency chain).
// A-layout (ISA 7.12.2): lanes 0-15 hold K={0,1} (2 VGPRs), lanes 16-31 K={2,3};
// B assumed row-striped like C (VGPR0: K0|K2, VGPR1: K1|K3) -> identical
// per-lane addressing for A and B: float2 at src + 4t + (lane<16 ? 0 : 2).
// ---------------------------------------------------------------------------
__global__ void cpm_main(const float* __restrict__ f_orig,
                         const float* __restrict__ f_gen,
                         const int*   __restrict__ pids,
                         const int*   __restrict__ camids,
                         const float* __restrict__ center_rgb,
                         const float* __restrict__ center_sar,
                         const float* __restrict__ wsamp,
                         float* __restrict__ wave_out) {
    __shared__ float cmat[8][8][32];            // 8 waves x 8 VGPRs x 32 lanes
    const int wave = threadIdx.x >> 5;
    const int lane = threadIdx.x & 31;
    const int wid  = blockIdx.x * 8 + wave;     // 0 .. B_*P_-1 (exactly 3072)
    const int b = wid / P_;
    const int p = wid % P_;

    const int  pid = pids[b];
    const bool rgb = (camids[b] == 0);
    const float wb = wsamp[b];

    const float* gbase = f_gen  + (((size_t)b * K_) * P_ + p) * D_;   // k-stride = P_*D_
    const float* ovec  = f_orig + ((size_t)b * P_ + p) * D_;
    const float* cvec  = (rgb ? center_sar : center_rgb) + ((size_t)pid * P_ + p) * D_;

    const int m    = lane & 15;
    const int dsub = (lane < 16) ? 0 : 2;

    // A operand source per lane (m>=10: dummy, contributes only to unused D rows)
    const float* rowsrc = ovec;
    if (m < 8)       rowsrc = gbase + (size_t)m * (P_ * D_);
    else if (m == 8) rowsrc = cvec;
    // B operand source per lane (m>=10: dummy, contributes only to unused D cols)
    const float* colsrc = ovec;
    if (m == 0)           colsrc = cvec;
    else if (m >= 2 && m < 10) colsrc = gbase + (size_t)(m - 2) * (P_ * D_);

    const float* rp = rowsrc + dsub;
    const float* cp = colsrc + dsub;

    v8f acc0 = {};
    v8f acc1 = {};
    #pragma unroll 4
    for (int t = 0; t < D_ / 8; ++t) {          // two K=4 chunks per iteration
        const v2f a0 = *(const v2f*)(rp);
        const v2f b0 = *(const v2f*)(cp);
        const v2f a1 = *(const v2f*)(rp + 4);
        const v2f b1 = *(const v2f*)(cp + 4);
        rp += 8;
        cp += 8;
        acc0 = __builtin_amdgcn_wmma_f32_16x16x4_f32(
                false, a0, false, b0, (short)0, acc0, false, false);
        acc1 = __builtin_amdgcn_wmma_f32_16x16x4_f32(
                false, a1, false, b1, (short)0, acc1, false, false);
    }

    #pragma unroll
    for (int r = 0; r < 8; ++r) cmat[wave][r][lane] = acc0[r] + acc1[r];
    __syncthreads();

    float h = 0.0f;
    if (lane < 8) {
        const int k = lane;
        const float gc = cmat[wave][k][0];       // D[k][0]
        const float go = cmat[wave][k][1];       // D[k][1]
        const float gg = cmat[wave][k][2 + k];   // D[k][2+k]
        const float cc = cmat[wave][0][16];      // D[8][0]
        const float oo = cmat[wave][1][17];      // D[9][1]
        float dp2 = gg - 2.0f * gc + cc; dp2 = dp2 > 0.f ? dp2 : 0.f;
        float ds2 = gg - 2.0f * go + oo; ds2 = ds2 > 0.f ? ds2 : 0.f;
        float hinge = sqrtf(dp2) - sqrtf(ds2) + MARGIN_;
        h = hinge > 0.f ? hinge : 0.f;
    }
    // wave reduction over k-lanes 0..7 (lanes 8..31 contribute exact zeros)
    h += __shfl_down(h, 4, 32);
    h += __shfl_down(h, 2, 32);
    h += __shfl_down(h, 1, 32);
    if (lane == 0) wave_out[wid] = h * wb;       // one write per wave: deterministic
}

// ---------------------------------------------------------------------------
// Kernel 5: deterministic final reduction -> scalar loss
// ---------------------------------------------------------------------------
__global__ void cpm_final(const float* __restrict__ wave_out,
                          const float* __restrict__ id_count,
                          float* __restrict__ out) {
    __shared__ float ps[P_];
    const int t = threadIdx.x;                   // blockDim.x == 32
    if (t < P_) {
        float s = 0.0f;
        for (int i = t; i < B_ * P_; i += P_) s += wave_out[i];  // wid = b*P_+p
        ps[t] = s;
    }
    __syncthreads();
    if (t == 0) {
        const float idc = *id_count;
        const float denom = idc > 1.f ? idc : 1.f;
        float s = 0.0f;
        for (int i = 0; i < P_; ++i) s += ps[i] / denom;
        out[0] = (idc > 0.f) ? s / (float)P_ : 0.0f;
    }
}

// ---------------------------------------------------------------------------
// Workspace layout (floats):
//   [0)            center_rgb  64*6*1024 = 393216
//   [393216)       center_sar  393216
//   [786432)       cnt_rgb     64
//   [786496)       cnt_sar     64
//   [786560)       wsamp       512
//   [787072)       id_count    1 (+7 pad)
//   [787080)       wave_out    3072
// total ~3.16 MB
// ---------------------------------------------------------------------------
extern "C" void kernel_launch(void* const* d_in, const int* in_sizes, int n_in,
                              void* d_out, int out_size, void* d_ws, size_t ws_size,
                              hipStream_t stream) {
    const float* f_orig = (const float*)d_in[0];
    const float* f_gen  = (const float*)d_in[1];
    const int*   pids   = (const int*)d_in[2];
    const int*   camids = (const int*)d_in[3];
    float* ws = (float*)d_ws;

    float* center_rgb = ws;
    float* center_sar = ws + 393216;
    float* cnt_rgb    = ws + 786432;
    float* cnt_sar    = ws + 786496;
    float* wsamp      = ws + 786560;
    float* id_count   = ws + 787072;
    float* wave_out   = ws + 787080;
    float* out        = (float*)d_out;

    cpm_counts <<<1, 64, 0, stream>>>(pids, camids, cnt_rgb, cnt_sar, id_count);
    cpm_centers<<<NIDS * P_, 256, 0, stream>>>(f_orig, pids, camids,
                                               cnt_rgb, cnt_sar, center_rgb, center_sar);
    cpm_weights<<<2, 256, 0, stream>>>(pids, camids, cnt_rgb, cnt_sar, wsamp);
    cpm_main   <<<(B_ * P_) / 8, 256, 0, stream>>>(f_orig, f_gen, pids, camids,
                                                   center_rgb, center_sar, wsamp, wave_out);
    cpm_final  <<<1, 32, 0, stream>>>(wave_out, id_count, out);
}